// FeatureSimilarityLoss_756
// MI455X (gfx1250) — compile-verified
//
#include <hip/hip_runtime.h>
#include <hip/hip_bf16.h>
#include <math.h>

// ---------------------------------------------------------------------------
// FeatureSimilarityLoss for MI455X (gfx1250, wave32, WMMA).
//
// sim = (fn fn^T) * 10 over N=9216, D=64  -> 10.9 GFLOP on 2.36 MB of data:
// compute-bound -> use V_WMMA_F32_16X16X4_F32 (fp32 in/out, K=4) to keep the
// precision the exp(10*dot) terms need.  Row max of sim is identically 1/T
// (diagonal self-sim = 10 = Cauchy-Schwarz bound), so a single streaming pass
// with exp(s - 10) reproduces the reference's max-subtracted softmax sums.
// ---------------------------------------------------------------------------

#define RES          96
#define N_PIX        (RES * RES)            // 9216
#define NUM_CLASSES  5
#define FEAT_DIM     64
#define INV_T        10.0f                  // 1 / TEMPERATURE
#define EPS_F        1e-8f

#define TILE              16
#define NTILES            (N_PIX / TILE)    // 576
#define WAVES_PER_BLOCK   8
#define ROWS_PER_BLOCK    (WAVES_PER_BLOCK * TILE)   // 128
#define ROW_BLOCKS        (N_PIX / ROWS_PER_BLOCK)   // 72
#define CSPLIT            8
#define CTILES_PER_SPLIT  (NTILES / CSPLIT)          // 72
#define BSTRIDE           68   // padded LDS row stride (floats): conflict-free, 272B = 17*16B

typedef __attribute__((ext_vector_type(2))) float v2f;
typedef __attribute__((ext_vector_type(8))) float v8f;

// ---------------------------------------------------------------------------
// Workspace zeroing: accum[2] (S,V) + den/num/cnt (3*N contiguous floats)
// ---------------------------------------------------------------------------
__global__ void zero_kernel(float* __restrict__ accum, float* __restrict__ arrs) {
    int i = blockIdx.x * blockDim.x + threadIdx.x;
    if (i < 2) accum[i] = 0.0f;
    if (i < 3 * N_PIX) arrs[i] = 0.0f;
}

// ---------------------------------------------------------------------------
// Prep: per-pixel argmax label + L2-normalized feature row.
// ---------------------------------------------------------------------------
__global__ __launch_bounds__(256) void prep_kernel(const float* __restrict__ gt_seg,
                                                   const float* __restrict__ features,
                                                   int*  __restrict__ labels,
                                                   float* __restrict__ fn) {
    int p = blockIdx.x * blockDim.x + threadIdx.x;
    if (p >= N_PIX) return;

    // argmax over class channel, first max wins (matches jnp.argmax)
    const float* g = gt_seg + p * NUM_CLASSES;
    int best = 0;
    float bv = g[0];
#pragma unroll
    for (int c = 1; c < NUM_CLASSES; ++c) {
        float v = g[c];
        if (v > bv) { bv = v; best = c; }
    }
    labels[p] = best;

    const float4* fr = (const float4*)(features + (size_t)p * FEAT_DIM);
    float4 buf[FEAT_DIM / 4];
    float ss = 0.0f;
#pragma unroll
    for (int i = 0; i < FEAT_DIM / 4; ++i) {
        float4 v = fr[i];
        buf[i] = v;
        ss += v.x * v.x + v.y * v.y + v.z * v.z + v.w * v.w;
    }
    float inv = 1.0f / fmaxf(sqrtf(ss), 1e-12f);
    float4* out = (float4*)(fn + (size_t)p * FEAT_DIM);
#pragma unroll
    for (int i = 0; i < FEAT_DIM / 4; ++i) {
        float4 v = buf[i];
        v.x *= inv; v.y *= inv; v.z *= inv; v.w *= inv;
        out[i] = v;
    }
}

// ---------------------------------------------------------------------------
// Main: each wave computes a 16-row strip of sim = fn fn^T / T via chained
// V_WMMA_F32_16X16X4_F32 (16 WMMAs cover K=64).  8 waves share each 16-col
// B tile through LDS.  Grid: (72 row-blocks) x (8 column splits).
//
// fp32 A-matrix 16x4 layout: lane holds row M = lane%16; VGPRs {0,1} hold
// K = {0,1} (lanes 0-15) or K = {2,3} (lanes 16-31) of each K-chunk.
// B 4x16 mirrors this with N = lane%16.  C/D: lane holds rows v+8*(lane/16),
// col = lane%16  ->  per-row reductions fold across lanes with xor 1/2/4/8.
// ---------------------------------------------------------------------------
__global__ __launch_bounds__(256) void simloss_kernel(const float* __restrict__ fn,
                                                      const int*   __restrict__ labels,
                                                      float* __restrict__ den_g,
                                                      float* __restrict__ num_g,
                                                      float* __restrict__ cnt_g) {
    __shared__ float btile[TILE * BSTRIDE];
    __shared__ int   clab_sh[TILE];

    const int lane = threadIdx.x & 31;
    const int wave = threadIdx.x >> 5;
    const int row0 = (blockIdx.x * WAVES_PER_BLOCK + wave) * TILE;

    const int lmod = lane & 15;
    const int lhi  = lane >> 4;       // 0: lanes 0-15, 1: lanes 16-31
    const int koff = 2 * lhi;         // K sub-offset within each K=4 chunk

    // Load full A strip once: 16 chunks of (16x4 fp32) = 32 VGPRs/lane.
    v2f a[FEAT_DIM / 4];
    {
        const float* arow = fn + (size_t)(row0 + lmod) * FEAT_DIM + koff;
#pragma unroll
        for (int k = 0; k < FEAT_DIM / 4; ++k)
            a[k] = *(const v2f*)(arow + 4 * k);
    }

    // Labels for the 8 C-rows this lane holds.
    int rlab[8];
#pragma unroll
    for (int v = 0; v < 8; ++v) rlab[v] = labels[row0 + v + 8 * lhi];

    float den[8], num[8], cnt[8];
#pragma unroll
    for (int v = 0; v < 8; ++v) { den[v] = 0.0f; num[v] = 0.0f; cnt[v] = 0.0f; }

    const int ct_beg = blockIdx.y * CTILES_PER_SPLIT;
    const int ct_end = ct_beg + CTILES_PER_SPLIT;

    for (int ct = ct_beg; ct < ct_end; ++ct) {
        const int col0 = ct * TILE;

        __syncthreads();
        {   // cooperative B-tile load: 16 rows x 64 floats, float4 per thread
            int t = threadIdx.x;
            const float4* src = (const float4*)(fn + (size_t)(col0 + (t >> 4)) * FEAT_DIM) + (t & 15);
            float4 v = *src;
            float* dst = btile + (t >> 4) * BSTRIDE + (t & 15) * 4;
            *(float4*)dst = v;
            if (t < TILE) clab_sh[t] = labels[col0 + t];
        }
        __syncthreads();

        // 16 chained fp32 WMMAs: C = A(16x64) * B(64x16)
        v8f c = {};
        const float* brow = btile + lmod * BSTRIDE + koff;
#pragma unroll
        for (int k = 0; k < FEAT_DIM / 4; ++k) {
            v2f b = *(const v2f*)(brow + 4 * k);
            c = __builtin_amdgcn_wmma_f32_16x16x4_f32(
                    /*neg_a=*/false, a[k], /*neg_b=*/false, b,
                    /*c_mod=*/(short)0, c, /*reuse_a=*/false, /*reuse_b=*/false);
        }

        // Streaming masked-softmax sums; row max == INV_T exactly (diagonal).
        const int colIdx = col0 + lmod;
        const int clab   = clab_sh[lmod];
#pragma unroll
        for (int v = 0; v < 8; ++v) {
            const int rowIdx = row0 + v + 8 * lhi;
            float s = c[v] * INV_T;
            float e = __expf(s - INV_T);
            float w = ((rlab[v] == clab) && (rowIdx != colIdx)) ? 1.0f : 0.0f;
            den[v] += e;
            num[v] += w * e;
            cnt[v] += w;
        }
    }

    // Fold the 16 columns held across each half-wave.
#pragma unroll
    for (int v = 0; v < 8; ++v) {
#pragma unroll
        for (int m = 1; m < 16; m <<= 1) {
            den[v] += __shfl_xor(den[v], m, 32);
            num[v] += __shfl_xor(num[v], m, 32);
            cnt[v] += __shfl_xor(cnt[v], m, 32);
        }
    }

    if (lmod == 0) {   // lanes 0 and 16: rows v and v+8
#pragma unroll
        for (int v = 0; v < 8; ++v) {
            const int row = row0 + v + 8 * lhi;
            atomicAdd(&den_g[row], den[v]);
            atomicAdd(&num_g[row], num[v]);
            atomicAdd(&cnt_g[row], cnt[v]);
        }
    }
}

// ---------------------------------------------------------------------------
// Per-row loss terms -> scalar accumulators.
// ---------------------------------------------------------------------------
__global__ __launch_bounds__(256) void rowfinal_kernel(const float* __restrict__ den,
                                                       const float* __restrict__ num,
                                                       const float* __restrict__ cnt,
                                                       float* __restrict__ accum) {
    int i = blockIdx.x * blockDim.x + threadIdx.x;
    if (i >= N_PIX) return;
    float term  = -__logf((num[i] + EPS_F) / (den[i] + EPS_F));
    float valid = (cnt[i] > 0.0f) ? 1.0f : 0.0f;
    atomicAdd(&accum[0], term);
    atomicAdd(&accum[1], valid);
}

// Reference applies valid mask AFTER .mean(): out = mean * V / (V + eps).
__global__ void scalar_kernel(const float* __restrict__ accum, float* __restrict__ out) {
    if (threadIdx.x == 0 && blockIdx.x == 0) {
        float S = accum[0], V = accum[1];
        out[0] = (S / (float)N_PIX) * (V / (V + EPS_F));
    }
}

// ---------------------------------------------------------------------------
extern "C" void kernel_launch(void* const* d_in, const int* in_sizes, int n_in,
                              void* d_out, int out_size, void* d_ws, size_t ws_size,
                              hipStream_t stream) {
    (void)in_sizes; (void)n_in; (void)out_size; (void)ws_size;

    const float* gt_seg   = (const float*)d_in[0];   // [96,96,5] f32
    const float* features = (const float*)d_in[1];   // [9216,64] f32
    float* out = (float*)d_out;

    // Workspace layout (offsets in bytes):
    //   0      : accum[2]           (S, V)
    //   256    : den[N] num[N] cnt[N]   (3*N contiguous floats)
    //   256+12N: labels[N] (int)
    //   256+16N: fn[N*64] (normalized features, f32)
    char* ws = (char*)d_ws;
    float* accum  = (float*)ws;
    float* arrs   = (float*)(ws + 256);
    float* den_g  = arrs;
    float* num_g  = arrs + N_PIX;
    float* cnt_g  = arrs + 2 * N_PIX;
    int*   labels = (int*)(ws + 256 + (size_t)12 * N_PIX);
    float* fn     = (float*)(ws + 256 + (size_t)16 * N_PIX);

    zero_kernel<<<(3 * N_PIX + 255) / 256, 256, 0, stream>>>(accum, arrs);
    prep_kernel<<<(N_PIX + 255) / 256, 256, 0, stream>>>(gt_seg, features, labels, fn);
    simloss_kernel<<<dim3(ROW_BLOCKS, CSPLIT), 256, 0, stream>>>(fn, labels, den_g, num_g, cnt_g);
    rowfinal_kernel<<<(N_PIX + 255) / 256, 256, 0, stream>>>(den_g, num_g, cnt_g, accum);
    scalar_kernel<<<1, 64, 0, stream>>>(accum, out);
}